// SOM_89498528514480
// MI455X (gfx1250) — compile-verified
//
#include <hip/hip_runtime.h>
#include <stdint.h>

typedef float v2f __attribute__((ext_vector_type(2)));
typedef float v4f __attribute__((ext_vector_type(4)));
typedef float v8f __attribute__((ext_vector_type(8)));

#define SOM_DIM 512
#define SOM_MN  65536
#define SOM_EPS 1e-6f
#define NB1     512           // pass-1 blocks: 512 blocks * 8 waves = 4096 tiles * 16 rows = 65536

// ---------------------------------------------------------------------------
// Pass 1: per-row squared distance via WMMA Gram-matrix diagonal + argmin keys
// ---------------------------------------------------------------------------
__global__ __launch_bounds__(256) void som_dist_kernel(
    const float* __restrict__ x, const float* __restrict__ W,
    unsigned long long* __restrict__ blockKeys)
{
    __shared__ float xe[SOM_DIM];            // x + EPS staged once per block
    __shared__ float cdump[8 * 256];         // per-wave C-matrix dump (8 waves)
    __shared__ unsigned long long wkeys[8];

    const int tid = threadIdx.x;
    xe[tid]       = x[tid]       + SOM_EPS;
    xe[tid + 256] = x[tid + 256] + SOM_EPS;
    __syncthreads();

    const int wave = tid >> 5;
    const int lane = tid & 31;
    const int half = lane >> 4;              // 0: K={0,1}, 1: K={2,3} per WMMA A layout
    const int lr16 = lane & 15;              // row within tile
    const int tile = blockIdx.x * 8 + wave;  // 0..4095
    const int row  = tile * 16 + lr16;       // 0..65535

    const float* wp = W  + (size_t)row * SOM_DIM + 2 * half;
    const float* xp = xe + 2 * half;

    v8f c = {0.f, 0.f, 0.f, 0.f, 0.f, 0.f, 0.f, 0.f};
    #pragma unroll 4
    for (int k0 = 0; k0 < SOM_DIM; k0 += 4) {
        v2f w2 = *(const v2f*)(wp + k0);
        v2f x2 = *(const v2f*)(xp + k0);
        v2f a;
        a.x = x2.x - w2.x;                   // (x+eps) - w ; squared == reference diff^2
        a.y = x2.y - w2.y;
        // B(4x16) layout is the transpose-mirror of A(16x4): same registers as
        // A and B -> D = A * A^T, diagonal = per-row sum of squares.
        c = __builtin_amdgcn_wmma_f32_16x16x4_f32(
                false, a, false, a, (short)0, c, false, false);
    }

    // dump accumulator, pull the diagonal: (i,i) -> vgpr i&7, lane i (i<8) / i+16
    float* cb = cdump + wave * 256;
    #pragma unroll
    for (int i = 0; i < 8; ++i) cb[i * 32 + lane] = c[i];
    __syncthreads();

    unsigned long long key = ~0ull;
    if (lane < 16) {
        float d2 = cb[(lane & 7) * 32 + (lane < 8 ? lane : lane + 16)];
        key = ((unsigned long long)__float_as_uint(d2) << 32) | (unsigned)row;
    }
    // wave32 min-reduce (d2 >= 0 so float bits order as uint; tie -> lowest row)
    #pragma unroll
    for (int off = 16; off > 0; off >>= 1) {
        unsigned long long o = __shfl_xor(key, off);
        if (o < key) key = o;
    }
    if (lane == 0) wkeys[wave] = key;
    __syncthreads();
    if (tid == 0) {
        unsigned long long m = wkeys[0];
        #pragma unroll
        for (int i = 1; i < 8; ++i) if (wkeys[i] < m) m = wkeys[i];
        blockKeys[blockIdx.x] = m;
    }
}

// ---------------------------------------------------------------------------
// Pass 2: reduce block keys -> BMU; compute decayed alpha and 1/sigma^2
// ---------------------------------------------------------------------------
__global__ __launch_bounds__(512) void som_bmu_kernel(
    const unsigned long long* __restrict__ blockKeys,
    const int* __restrict__ loc, const int* __restrict__ itp,
    float* __restrict__ params)
{
    __shared__ unsigned long long s[NB1];
    const int tid = threadIdx.x;
    s[tid] = blockKeys[tid];
    __syncthreads();
    for (int off = NB1 / 2; off > 0; off >>= 1) {
        if (tid < off) { if (s[tid + off] < s[tid]) s[tid] = s[tid + off]; }
        __syncthreads();
    }
    if (tid == 0) {
        unsigned bmu = (unsigned)(s[0] & 0xffffffffull);
        float bx = (float)loc[2 * bmu];
        float by = (float)loc[2 * bmu + 1];
        float decay = 1.0f - (float)itp[0] / 100.0f;   // NITER = 100
        float alpha = 0.3f * decay;                    // ALPHA = 0.3
        float sigma = 128.0f * decay;                  // SIGMA = max(M,N)/2
        params[0] = bx;
        params[1] = by;
        params[2] = alpha;
        params[3] = 1.0f / (sigma * sigma);
    }
}

// ---------------------------------------------------------------------------
// Pass 3: W' = W + lr*(x - W); W read is L2-hot, output stored non-temporally
// ---------------------------------------------------------------------------
__global__ __launch_bounds__(256) void som_update_kernel(
    const float* __restrict__ x, const float* __restrict__ W,
    const int* __restrict__ loc, const float* __restrict__ params,
    float* __restrict__ out)
{
    const int gid = blockIdx.x * 256 + threadIdx.x;   // one float4 per thread
    const int row = gid >> 7;                         // 128 float4 per row
    const int c4  = gid & 127;

    const float bx = params[0], by = params[1];
    const float alpha = params[2], inv_s2 = params[3];

    const float dx = (float)loc[2 * row]     - bx;
    const float dy = (float)loc[2 * row + 1] - by;
    const float lr = alpha * __expf(-(dx * dx + dy * dy) * inv_s2);

    const size_t off = (size_t)row * SOM_DIM + (size_t)c4 * 4;
    v4f w4 = *(const v4f*)(W + off);
    v4f x4 = *(const v4f*)(x + c4 * 4);
    v4f o  = w4 + lr * (x4 - w4);
    __builtin_nontemporal_store(o, (v4f*)(out + off));
}

// ---------------------------------------------------------------------------
extern "C" void kernel_launch(void* const* d_in, const int* in_sizes, int n_in,
                              void* d_out, int out_size, void* d_ws, size_t ws_size,
                              hipStream_t stream)
{
    const float* x   = (const float*)d_in[0];   // [512]
    const float* W   = (const float*)d_in[1];   // [65536*512]
    const int*   loc = (const int*)d_in[2];     // [65536*2]
    const int*   itp = (const int*)d_in[3];     // [1]
    float*       out = (float*)d_out;           // [65536*512]

    unsigned long long* keys   = (unsigned long long*)d_ws;            // NB1 * 8 B
    float*              params = (float*)((char*)d_ws + NB1 * 8);      // 4 floats

    som_dist_kernel  <<<NB1,                     256, 0, stream>>>(x, W, keys);
    som_bmu_kernel   <<<1,                       NB1, 0, stream>>>(keys, loc, itp, params);
    som_update_kernel<<<(SOM_MN * (SOM_DIM/4))/256, 256, 0, stream>>>(x, W, loc, params, out);
}